// AdaFace_64132451664174
// MI455X (gfx1250) — compile-verified
//
#include <hip/hip_runtime.h>
#include <math.h>

typedef float v2f __attribute__((ext_vector_type(2)));
typedef float v8f __attribute__((ext_vector_type(8)));
typedef int   v4i __attribute__((vector_size(16)));

#define BN   512          // batch rows (M)
#define DD   512          // embedding dim (K)
#define CC   70722        // classes (N)
#define SVAL 64.0f
#define MMAR 0.4f
#define HVAL 0.333f
#define EPSV 1e-3f
#define PI_F 3.14159265358979323846f

#define KB   32           // K chunk per LDS stage
#define NCH  (DD / KB)    // 16 chunks

#define AS1 __attribute__((address_space(1)))
#define AS3 __attribute__((address_space(3)))

#if __has_builtin(__builtin_amdgcn_global_load_async_to_lds_b32) && \
    __has_builtin(__builtin_amdgcn_global_load_async_to_lds_b128)
#define USE_ASYNC 1
#else
#define USE_ASYNC 0
#endif

static __device__ __forceinline__ void wait_async0() {
#if __has_builtin(__builtin_amdgcn_s_wait_asynccnt)
    __builtin_amdgcn_s_wait_asynccnt(0);
#else
    asm volatile("s_wait_asynccnt 0x0" ::: "memory");
#endif
}

#if USE_ASYNC
static __device__ __forceinline__ void async_b32(const float* g, float* l) {
    __builtin_amdgcn_global_load_async_to_lds_b32((AS1 int*)g, (AS3 int*)l, 0, 0);
}
static __device__ __forceinline__ void async_b128(const float* g, float* l) {
    __builtin_amdgcn_global_load_async_to_lds_b128((AS1 v4i*)g, (AS3 v4i*)l, 0, 0);
}
#endif

// ---------------- column inverse norms of kernel [D, C] ----------------
__global__ __launch_bounds__(256)
void colnorm_kernel(const float* __restrict__ K, float* __restrict__ invk) {
    int j = blockIdx.x * 256 + threadIdx.x;
    if (j >= CC) return;
    float s = 0.0f;
    for (int d = 0; d < DD; ++d) {
        float v = K[(size_t)d * CC + j];   // coalesced across threads
        s += v * v;
    }
    invk[j] = 1.0f / sqrtf(s);
}

// ---------------- row norms of x [B, D] ----------------
__global__ __launch_bounds__(64)
void rownorm_kernel(const float* __restrict__ X, float* __restrict__ nrm,
                    float* __restrict__ invx) {
    __shared__ float red[64];
    int i = blockIdx.x, t = threadIdx.x;
    float s = 0.0f;
    for (int j = t; j < DD; j += 64) {
        float v = X[(size_t)i * DD + j];
        s += v * v;
    }
    red[t] = s; __syncthreads();
    for (int off = 32; off > 0; off >>= 1) {
        if (t < off) red[t] += red[t + off];
        __syncthreads();
    }
    if (t == 0) {
        float n = sqrtf(red[0]);
        nrm[i] = n;
        invx[i] = 1.0f / n;
    }
}

// ---------------- margin statistics (single block of 512) ----------------
__global__ __launch_bounds__(512)
void stats_kernel(const float* __restrict__ nrm, float* __restrict__ g_ang,
                  float* __restrict__ g_add) {
    __shared__ float red[512];
    int t = threadIdx.x;
    float v = nrm[t];
    v = fminf(fmaxf(v, 1e-3f), 100.0f);          // safe_norms
    red[t] = v; __syncthreads();
    for (int off = 256; off > 0; off >>= 1) {
        if (t < off) red[t] += red[t + off];
        __syncthreads();
    }
    float mean = red[0] * (1.0f / 512.0f);
    __syncthreads();
    float d = v - mean;
    red[t] = d * d; __syncthreads();
    for (int off = 256; off > 0; off >>= 1) {
        if (t < off) red[t] += red[t + off];
        __syncthreads();
    }
    float stdv = sqrtf(red[0] / 511.0f);         // unbiased (ddof=1)
    float ms = d / (stdv + EPSV) * HVAL;
    ms = fminf(fmaxf(ms, -1.0f), 1.0f);
    g_ang[t] = -MMAR * ms;
    g_add[t] = MMAR + MMAR * ms;
}

// ---------------- fp32 WMMA GEMM: out = S * clip(invx .* (X@K) .* invk) ----
// block = 256 thr = 8 waves, 4(M) x 2(N); wave tile 16x128; block tile 64x256
// double-buffered LDS, KB=32 K-chunks, async global->LDS when available
__global__ __launch_bounds__(256)
void gemm_wmma_kernel(const float* __restrict__ X, const float* __restrict__ K,
                      const float* __restrict__ invx, const float* __restrict__ invk,
                      float* __restrict__ out) {
    __shared__ float As[2][64][KB];          // [buf][m][k] row-major, 2 x 8 KB
    __shared__ float Bs[2][KB / 4][256][4];  // [buf][k/4][n][k%4] swizzled, 2 x 32 KB

    const int tid   = threadIdx.x;
    const int lane  = tid & 31;
    const int wid   = tid >> 5;
    const int wm    = wid & 3;            // wave M position (0..3)
    const int wn    = wid >> 2;           // wave N position (0..1)
    const int l15   = lane & 15;
    const int lhalf = lane >> 4;          // 0 | 1

    const int mb = blockIdx.y;            // 0..7   (M blocks of 64)
    const int nb = blockIdx.x;            // 0..276 (N blocks of 256)
    const int n_base = nb * 256;

    v8f acc[8];
    const v8f vzero = {0.f,0.f,0.f,0.f,0.f,0.f,0.f,0.f};
#pragma unroll
    for (int t = 0; t < 8; ++t) acc[t] = vzero;

    // staging indices (loop-invariant)
    const int a_m  = tid >> 2;                   // 0..63
    const int a_c8 = (tid & 3) << 3;             // 0,8,16,24
    const int nglob = n_base + tid;
    const int nclmp = nglob < CC ? nglob : CC - 1;
    const float* aSrcBase = X + (size_t)(mb * 64 + a_m) * DD + a_c8;
    const float* bSrcBase = K + nclmp;

#if USE_ASYNC
    // ---- async global->LDS staging (ASYNCcnt-tracked, no VGPR round trip) ----
#define STAGE(buf, k0)                                                          \
    do {                                                                        \
        async_b128(aSrcBase + (k0),     &As[(buf)][a_m][a_c8]);                 \
        async_b128(aSrcBase + (k0) + 4, &As[(buf)][a_m][a_c8 + 4]);             \
        _Pragma("unroll")                                                       \
        for (int r = 0; r < KB; ++r) {                                          \
            async_b32(bSrcBase + (size_t)((k0) + r) * CC,                       \
                      &Bs[(buf)][r >> 2][tid][r & 3]);                          \
        }                                                                       \
    } while (0)

    STAGE(0, 0);
    for (int c = 0; c < NCH; ++c) {
        const int cur = c & 1;
        wait_async0();                 // own chunk-c writes complete
        __syncthreads();               // everyone's writes visible, compute c-1 done
        if (c + 1 < NCH) STAGE(cur ^ 1, (c + 1) * KB);   // fly during compute
#else
    // ---- fallback: register prefetch + double-buffered LDS ----
    float4 aReg[2];
    float  bReg[KB];
#define LOADC(k0)                                                               \
    do {                                                                        \
        aReg[0] = *(const float4*)(aSrcBase + (k0));                            \
        aReg[1] = *(const float4*)(aSrcBase + (k0) + 4);                        \
        _Pragma("unroll")                                                       \
        for (int r = 0; r < KB; ++r)                                            \
            bReg[r] = bSrcBase[(size_t)((k0) + r) * CC];                        \
    } while (0)

    LOADC(0);
    for (int c = 0; c < NCH; ++c) {
        const int cur = c & 1;
        *(float4*)(&As[cur][a_m][a_c8])     = aReg[0];
        *(float4*)(&As[cur][a_m][a_c8 + 4]) = aReg[1];
#pragma unroll
        for (int r = 0; r < KB; ++r) Bs[cur][r >> 2][tid][r & 3] = bReg[r];
        if (c + 1 < NCH) LOADC((c + 1) * KB);            // fly during compute
        __syncthreads();               // stores visible; compute c-1 globally done
#endif

        // ---- compute chunk c from Bs/As[cur] ----
#pragma unroll
        for (int kk = 0; kk < KB / 4; ++kk) {
            // A fragment 16x4: lanes 0-15 -> K={0,1}, lanes 16-31 -> K={2,3}
            v2f a = *(const v2f*)(&As[cur][wm * 16 + l15][kk * 4 + lhalf * 2]);
#pragma unroll
            for (int t = 0; t < 8; ++t) {
                // B fragment 4x16: v0 row = lhalf, v1 row = lhalf+2
                const float* bp = &Bs[cur][kk][wn * 128 + t * 16 + l15][lhalf];
                v2f b;
                b.x = bp[0];
                b.y = bp[2];                  // 8 B apart -> ds_load_2addr_b32
                acc[t] = __builtin_amdgcn_wmma_f32_16x16x4_f32(
                    false, a, false, b, (short)0, acc[t], false, false);
            }
        }
        __syncthreads();  // reads of buf[cur] done before it is rewritten at c+2
    }

    // ---- epilogue: scale by inv norms, clip, *S, predicated store ----
#pragma unroll
    for (int t = 0; t < 8; ++t) {
        int n = n_base + wn * 128 + t * 16 + l15;
        bool ok = n < CC;
        float ik = invk[ok ? n : CC - 1];
#pragma unroll
        for (int r = 0; r < 8; ++r) {
            int m = mb * 64 + wm * 16 + lhalf * 8 + r;   // D-matrix VGPR layout
            float c = acc[t][r] * invx[m] * ik;
            c = fminf(fmaxf(c, -1.0f + EPSV), 1.0f - EPSV);
            if (ok) out[(size_t)m * CC + n] = c * SVAL;
        }
    }
}

// ---------------- label-column margin fixup ----------------
__global__ __launch_bounds__(256)
void fixup_kernel(const long long* __restrict__ label, const float* __restrict__ g_ang,
                  const float* __restrict__ g_add, float* __restrict__ out) {
    int i = blockIdx.x * blockDim.x + threadIdx.x;
    if (i >= BN) return;
    long long j = label[i];
    size_t idx = (size_t)i * CC + (size_t)j;
    float c = out[idx] * (1.0f / SVAL);          // clipped cosine
    float th = acosf(c) + g_ang[i];
    th = fminf(fmaxf(th, EPSV), PI_F - EPSV);
    out[idx] = (cosf(th) - g_add[i]) * SVAL;
}

extern "C" void kernel_launch(void* const* d_in, const int* in_sizes, int n_in,
                              void* d_out, int out_size, void* d_ws, size_t ws_size,
                              hipStream_t stream) {
    const float*     X     = (const float*)d_in[0];      // [512, 512]
    const long long* label = (const long long*)d_in[1];  // [512] int64
    const float*     K     = (const float*)d_in[2];      // [512, 70722]
    float*           out   = (float*)d_out;              // [512, 70722]

    float* invk  = (float*)d_ws;          // C floats (padded to 70784)
    float* nrm   = invk + 70784;          // 512
    float* invx  = nrm + 512;             // 512
    float* g_ang = invx + 512;            // 512
    float* g_add = g_ang + 512;           // 512

    colnorm_kernel<<<dim3((CC + 255) / 256), 256, 0, stream>>>(K, invk);
    rownorm_kernel<<<dim3(BN), 64, 0, stream>>>(X, nrm, invx);
    stats_kernel<<<dim3(1), 512, 0, stream>>>(nrm, g_ang, g_add);
    gemm_wmma_kernel<<<dim3(277, 8), 256, 0, stream>>>(X, K, invx, invk, out);
    fixup_kernel<<<dim3(2), 256, 0, stream>>>(label, g_ang, g_add, out);
}